// ManifoldHyperConnectionFuse_5136780886166
// MI455X (gfx1250) — compile-verified
//
#include <hip/hip_runtime.h>
#include <stdint.h>

typedef float v2f __attribute__((ext_vector_type(2)));
typedef float v4f __attribute__((ext_vector_type(4)));
typedef float v8f __attribute__((ext_vector_type(8)));
typedef unsigned int u32x4 __attribute__((ext_vector_type(4)));
typedef int i32x4 __attribute__((ext_vector_type(4)));
typedef int i32x8 __attribute__((ext_vector_type(8)));

#define TOK_PER_BLK 16
#define THREADS     128
#define HSTRIDE     1028   // 1024 + 4 pad floats -> conflict-free A-fragment reads

__launch_bounds__(THREADS)
__global__ void mhc_fused_kernel(const float* __restrict__ h,
                                 const float* __restrict__ nw,
                                 const float* __restrict__ w,
                                 const float* __restrict__ alpha,
                                 const float* __restrict__ beta,
                                 float* __restrict__ out)
{
    __shared__ float s_h[TOK_PER_BLK * HSTRIDE];       // 65792 B raw h tile
    __shared__ float s_nw[1024];                       // 4 KB norm_weight
    __shared__ float s_ssq[TOK_PER_BLK];               // per-token sum of squares
    __shared__ float s_Hpart[2][2][TOK_PER_BLK][16];   // [ntile][khalf][token][n]
    __shared__ float s_M[TOK_PER_BLK][16];             // fused 4x4 per token

    const int tid  = threadIdx.x;
    const int lane = tid & 31;
    const int wv   = tid >> 5;
    const long bt  = (long)blockIdx.x * TOK_PER_BLK;   // first token of block
    const float* hblk = h + bt * 1024;

    if (tid < TOK_PER_BLK) s_ssq[tid] = 0.0f;
    // stage norm_weight (1024 floats, 2 float4 per thread)
    {
        const v4f* nw4 = (const v4f*)nw;
        v4f* snw4 = (v4f*)s_nw;
        snw4[tid]       = nw4[tid];
        snw4[tid + 128] = nw4[tid + 128];
    }

    // ---- Phase 1: TDM async tensor load: each wave DMAs 4 token rows ----
    // Iterate mode: per iteration copy one 1024-f32 row, stepping LDS by the
    // padded stride (1028 floats) and global by 1024 floats; 4 iterations.
    {
        const float*  gbase    = hblk + (size_t)wv * 4 * 1024;
        const uint64_t ga      = (uint64_t)(uintptr_t)gbase;
        const uint32_t lds_off = (uint32_t)(uintptr_t)(void*)(s_h + wv * 4 * HSTRIDE);

        u32x4 g0;
        g0.x = 1u;                                   // count=1 (valid), user mode
        g0.y = lds_off;                              // lds_addr (bytes)
        g0.z = (uint32_t)ga;                         // global_addr[31:0]
        g0.w = (uint32_t)((ga >> 32) & 0x01FFFFFFu)  // global_addr[56:32]
             | (2u << 30);                           // type = 2 ("image")

        i32x8 g1;
        g1[0] = (2 << 16) | (1 << 19);   // data_size=4B, iterate_enable=1
        g1[1] = (int)(1024u << 16);      // tensor_dim0 = 1024 (bits 79:48)
        g1[2] = (int)(1u << 16);         // tensor_dim1 = 1    (bits 111:80)
        g1[3] = (int)(1024u << 16);      // tile_dim0  = 1024  (bits 127:112)
        g1[4] = 1;                       // tile_dim1 = 1, tile_dim2 = 0
        g1[5] = 1024;                    // tensor_dim0_stride = 1024
        g1[6] = 0;                       // dim0_stride hi / dim1_stride lo
        g1[7] = 0;                       // tensor_dim1_stride hi

        i32x4 g2;                        // iterate-mode group 2
        g2.x = 1;                        // tensor_dim2 (unused w/ iterate)
        g2.y = 1028;                     // lds_addr_increment (data_size units)
        g2.z = 1024;                     // global_addr_increment (data_size units)
        g2.w = (int)(3u << 16);          // iterate_count = 3 -> 4 iterations

        i32x4 g3 = {0, 0, 0, 0};

#if defined(__clang_major__) && __clang_major__ >= 23
        i32x8 g4 = {0, 0, 0, 0, 0, 0, 0, 0};
        __builtin_amdgcn_tensor_load_to_lds(g0, g1, g2, g3, g4, 0);
#else
        __builtin_amdgcn_tensor_load_to_lds(g0, g1, g2, g3, 0);
#endif
        __builtin_amdgcn_s_wait_tensorcnt(0);
    }
    __syncthreads();

    // ---- Phase 1b: per-token sum of squares from the LDS tile ----
    for (int T = 0; T < TOK_PER_BLK; ++T) {
        float s = 0.0f;
        #pragma unroll
        for (int half = 0; half < 2; ++half) {
            int c4 = half * THREADS + tid;               // float4 col in [0,256)
            v4f gv = ((const v4f*)(s_h + T * HSTRIDE))[c4];
            s += gv.x * gv.x + gv.y * gv.y + gv.z * gv.z + gv.w * gv.w;
        }
        #pragma unroll
        for (int off = 16; off > 0; off >>= 1) s += __shfl_xor(s, off, 32);
        if (lane == 0) atomicAdd(&s_ssq[T], s);
    }
    __syncthreads();

    // ---- Phase 2: H = (nw * hf) @ w via V_WMMA_F32_16X16X4_F32 ----
    const int t    = wv & 1;          // n-tile: cols [t*16, t*16+16)
    const int kh   = wv >> 1;         // K half: [kh*512, kh*512+512)
    const int m    = lane & 15;       // A row (token within tile)
    const int hh   = lane >> 4;       // 0: K={0,1}, 1: K={2,3}
    const int n16  = lane & 15;       // B/C column within tile
    const int ncol = (t * 16 + n16 > 23) ? 23 : (t * 16 + n16);  // clamp

    v8f c = {0.f, 0.f, 0.f, 0.f, 0.f, 0.f, 0.f, 0.f};
    const float* srow = s_h + m * HSTRIDE;
    const float* wl   = w + ncol;
    const int k0 = __builtin_amdgcn_readfirstlane(kh * 512);  // wave-uniform
    #pragma unroll 4
    for (int k = k0; k < k0 + 512; k += 4) {
        const int kk = k + 2 * hh;
        v2f a, b;
        a.x = srow[kk]     * s_nw[kk];       // A: hv[m][kk], hv[m][kk+1]
        a.y = srow[kk + 1] * s_nw[kk + 1];
        b.x = wl[kk * 24];                   // B: w[kk][ncol], w[kk+1][ncol]
        b.y = wl[(kk + 1) * 24];
        c = __builtin_amdgcn_wmma_f32_16x16x4_f32(
                /*neg_a=*/false, a, /*neg_b=*/false, b,
                /*c_mod=*/(short)0, c, /*reuse_a=*/false, /*reuse_b=*/false);
    }
    // scatter C tile: vgpr j -> token j (lanes 0-15) / j+8 (lanes 16-31)
    #pragma unroll
    for (int j = 0; j < 8; ++j) {
        s_Hpart[t][kh][j + 8 * hh][n16] = c[j];
    }
    __syncthreads();

    // ---- Phase 3: per-token scalar chain (one thread per token) ----
    if (tid < TOK_PER_BLK) {
        float H[24];
        #pragma unroll
        for (int j = 0; j < 16; ++j)
            H[j] = s_Hpart[0][0][tid][j] + s_Hpart[0][1][tid][j];
        #pragma unroll
        for (int j = 0; j < 8; ++j)
            H[16 + j] = s_Hpart[1][0][tid][j] + s_Hpart[1][1][tid][j];

        const float a0 = alpha[0], a1 = alpha[1], a2 = alpha[2];
        const float r  = sqrtf(s_ssq[tid]) * (1.0f / 32.0f);   // /sqrt(1024)
        const float r_ = 1.0f / (r + 1e-6f);

        float Hpre[4], Hpost[4];
        #pragma unroll
        for (int i = 0; i < 4; ++i) {
            Hpre[i]  = 1.0f / (1.0f + __expf(-(r_ * H[i]     * a0 + beta[i])));
            Hpost[i] = 2.0f / (1.0f + __expf(-(r_ * H[4 + i] * a1 + beta[4 + i])));
        }
        float Km[4][4];
        #pragma unroll
        for (int i = 0; i < 4; ++i)
            #pragma unroll
            for (int j = 0; j < 4; ++j)
                Km[i][j] = __expf(r_ * H[8 + i * 4 + j] * a2 + beta[8 + i * 4 + j]);

        // Sinkhorn-Knopp, 10 iterations (u uses old v; v uses new u)
        float u[4] = {1.f, 1.f, 1.f, 1.f}, v[4] = {1.f, 1.f, 1.f, 1.f};
        for (int it = 0; it < 10; ++it) {
            #pragma unroll
            for (int i = 0; i < 4; ++i) {
                float s = Km[i][0]*v[0] + Km[i][1]*v[1] + Km[i][2]*v[2] + Km[i][3]*v[3];
                u[i] = 1.0f / (s + 1e-8f);
            }
            #pragma unroll
            for (int j = 0; j < 4; ++j) {
                float s = Km[0][j]*u[0] + Km[1][j]*u[1] + Km[2][j]*u[2] + Km[3][j]*u[3];
                v[j] = 1.0f / (s + 1e-8f);
            }
        }
        // fused coefficient matrix: M[n][m] = Hpost[n]*Hpre[m] + P[n][m]
        #pragma unroll
        for (int i = 0; i < 4; ++i)
            #pragma unroll
            for (int j = 0; j < 4; ++j)
                s_M[tid][i * 4 + j] = Hpost[i] * Hpre[j] + u[i] * Km[i][j] * v[j];
    }
    __syncthreads();

    // ---- Phase 4: out[token][n][d] = sum_m M[n][m] * h[token][m][d] ----
    v4f* oblk = (v4f*)(out + bt * 1024);
    #pragma unroll 4
    for (int iter = 0; iter < 32; ++iter) {
        int idx   = iter * THREADS + tid;   // float4 index in [0,4096)
        int token = idx >> 8;
        int rem   = idx & 255;
        int n     = rem >> 6;
        int d4    = rem & 63;
        const float* hr = s_h + token * HSTRIDE;
        v4f acc = {0.f, 0.f, 0.f, 0.f};
        #pragma unroll
        for (int mm = 0; mm < 4; ++mm) {
            float coef = s_M[token][n * 4 + mm];
            v4f hv4 = ((const v4f*)(hr + mm * 256))[d4];
            acc += coef * hv4;
        }
        __builtin_nontemporal_store(acc, &oblk[idx]);  // write-once stream
    }
}

extern "C" void kernel_launch(void* const* d_in, const int* in_sizes, int n_in,
                              void* d_out, int out_size, void* d_ws, size_t ws_size,
                              hipStream_t stream) {
    const float* h     = (const float*)d_in[0];   // (8,4096,4,256)
    const float* nw    = (const float*)d_in[1];   // (1024,)
    const float* w     = (const float*)d_in[2];   // (1024,24)
    const float* alpha = (const float*)d_in[3];   // (3,)
    const float* beta  = (const float*)d_in[4];   // (24,)
    float* out = (float*)d_out;                   // (8,4096,4,256)

    const int total_tokens = in_sizes[0] / 1024;  // 32768
    dim3 grid(total_tokens / TOK_PER_BLK);        // 2048 blocks
    dim3 block(THREADS);                          // 4 waves
    mhc_fused_kernel<<<grid, block, 0, stream>>>(h, nw, w, alpha, beta, out);
}